// DecoderSVM_36541581754605
// MI455X (gfx1250) — compile-verified
//
#include <hip/hip_runtime.h>

typedef float v2f __attribute__((ext_vector_type(2)));
typedef float v8f __attribute__((ext_vector_type(8)));

#define B_   256
#define I_   182
#define T_   2000
#define O_   2
#define WROW 188      // padded W row length in LDS floats (even, >= 184)
#define NROW 16       // pad B-matrix to all 16 N rows (rows >= O_ are zero)
#define NT   125      // number of 16-wide t tiles (125*16 == 2000)
#define NC   10       // scan chunks
#define CH   200      // chunk length (NC*CH == T_)

// ---------------- Pass 1: curr[b,o,t] = sum_i inputs[b,i,t] * W[o,i] ----------------
// One wave per (b, 16-t tile). M = t (16), N = o-slots (16, 2 live), K = i.
// V_WMMA_F32_16X16X4_F32, K stepped by 4. Inner loop is branch/divergence free.
__global__ __launch_bounds__(128) void snn_gemm_wmma(
    const float* __restrict__ inp,   // [B, I, T]
    const float* __restrict__ W,     // [O, I]
    float* __restrict__ curr)        // [B, O, T]
{
    __shared__ float lds_w[NROW * WROW];
    for (int idx = threadIdx.x; idx < NROW * WROW; idx += blockDim.x) {
        int o = idx / WROW, i = idx % WROW;
        lds_w[idx] = (o < O_ && i < I_) ? W[o * I_ + i] : 0.0f;
    }
    __syncthreads();

    const int b    = blockIdx.x;
    const int wave = threadIdx.x >> 5;
    const int tile = blockIdx.y * 4 + wave;
    if (tile >= NT) return;                 // uniform per wave: EXEC all-1s for WMMA

    const int lane  = threadIdx.x & 31;
    const int row   = lane & 15;            // A: M (t_local); B/C/D: N (o slot)
    const int khalf = lane >> 4;            // selects K pair within the fragment
    const int t0    = tile * 16;

    const size_t basebt = (size_t)b * I_ * T_ + (size_t)(t0 + row);
    const float2* __restrict__ wrow = (const float2*)&lds_w[row * WROW];

    v8f acc = {};
    // 45 full K-steps: i = k + 2*khalf + {0,1} <= 179 < I_, no bounds checks needed.
    for (int k = 0; k < 180; k += 4) {
        const int i0 = k + 2 * khalf;
        float a0 = inp[basebt + (size_t)i0 * T_];
        float a1 = inp[basebt + (size_t)(i0 + 1) * T_];
        float2 w = wrow[(k >> 1) + khalf];          // ds_load_b64, unconditional
        v2f Af = {a0, a1};
        v2f Bf = {w.x, w.y};
        acc = __builtin_amdgcn_wmma_f32_16x16x4_f32(
            false, Af, false, Bf, (short)0, acc, false, false);
    }
    // Tail K-step covers i = 180..183; khalf==1 lanes (i=182,183) masked to zero
    // via multiply with a clamped in-bounds address (no exec divergence, no OOB).
    {
        float m  = (khalf == 0) ? 1.0f : 0.0f;
        int   ia = (khalf == 0) ? 180 : 0;          // safe address, value masked
        float a0 = m * inp[basebt + (size_t)ia * T_];
        float a1 = m * inp[basebt + (size_t)(ia + 1) * T_];
        float2 w = wrow[90 + khalf];                // lds_w zero-padded to 184+
        v2f Af = {a0, a1};
        v2f Bf = {w.x, w.y};
        acc = __builtin_amdgcn_wmma_f32_16x16x4_f32(
            false, Af, false, Bf, (short)0, acc, false, false);
    }

    // C/D layout: VGPR r -> M = r + 8*khalf, N = lane&15. Keep only N < O_.
    if (row < O_) {
        size_t base = ((size_t)b * O_ + (size_t)row) * T_ + (size_t)(t0 + khalf * 8);
#pragma unroll
        for (int r = 0; r < 8; ++r) curr[base + r] = acc[r];
    }
}

__device__ __forceinline__ float clamp01(float x) {
    return fminf(fmaxf(x, 0.0f), 1.0f);
}

// ---------------- Pass 2a: per-chunk local scan (zero init), both o per thread -----
__global__ void snn_chunk_local(const float* __restrict__ curr,
                                const float* __restrict__ bias,
                                const float* __restrict__ alpha,
                                const float* __restrict__ beta,
                                float* __restrict__ summ)   // [B*O*NC][2]
{
    int tid = blockIdx.x * blockDim.x + threadIdx.x;
    if (tid >= B_ * NC) return;
    int c = tid % NC;
    int b = tid / NC;
    float a0 = clamp01(alpha[0]), a1 = clamp01(alpha[1]);
    float b0 = clamp01(beta[0]),  b1 = clamp01(beta[1]);
    float s0 = bias[0],           s1 = bias[1];
    const float4* r0 = (const float4*)&curr[((size_t)b * O_ + 0) * T_ + (size_t)c * CH];
    const float4* r1 = (const float4*)&curr[((size_t)b * O_ + 1) * T_ + (size_t)c * CH];
    float syn0 = 0.f, mem0 = 0.f, syn1 = 0.f, mem1 = 0.f;
    for (int j = 0; j < CH / 4; ++j) {
        float4 c0 = r0[j], c1 = r1[j];
        syn0 = a0 * syn0 + (c0.x + s0); mem0 = b0 * mem0 + syn0;
        syn1 = a1 * syn1 + (c1.x + s1); mem1 = b1 * mem1 + syn1;
        syn0 = a0 * syn0 + (c0.y + s0); mem0 = b0 * mem0 + syn0;
        syn1 = a1 * syn1 + (c1.y + s1); mem1 = b1 * mem1 + syn1;
        syn0 = a0 * syn0 + (c0.z + s0); mem0 = b0 * mem0 + syn0;
        syn1 = a1 * syn1 + (c1.z + s1); mem1 = b1 * mem1 + syn1;
        syn0 = a0 * syn0 + (c0.w + s0); mem0 = b0 * mem0 + syn0;
        syn1 = a1 * syn1 + (c1.w + s1); mem1 = b1 * mem1 + syn1;
    }
    int base0 = ((b * O_ + 0) * NC + c) * 2;
    int base1 = ((b * O_ + 1) * NC + c) * 2;
    summ[base0 + 0] = syn0; summ[base0 + 1] = mem0;
    summ[base1 + 0] = syn1; summ[base1 + 1] = mem1;
}

// ---------------- Pass 2b: combine chunk summaries -> per-chunk initial states ------
// x_end = M^CH * x0 + local, M = [[a,0],[a,b]]
// M^CH = [[aL,0],[uL,bL]], aL=a^CH, bL=b^CH, uL = a*(aL-bL)/(a-b) (CH*aL if a==b)
__global__ void snn_chunk_scan(const float* __restrict__ summ,
                               const float* __restrict__ alpha,
                               const float* __restrict__ beta,
                               float* __restrict__ inits)   // [B*O*NC][2]
{
    int bo = blockIdx.x * blockDim.x + threadIdx.x;
    if (bo >= B_ * O_) return;
    int o = bo % O_;
    float a  = clamp01(alpha[o]);
    float bt = clamp01(beta[o]);
    float aL = 1.0f, bL = 1.0f;
    for (int j = 0; j < CH; ++j) { aL *= a; bL *= bt; }
    float d  = a - bt;
    float uL = (fabsf(d) > 1e-5f) ? a * (aL - bL) / d : (float)CH * aL;
    float syn = 0.0f, mem = 0.0f;
    for (int c = 0; c < NC; ++c) {
        int idx = (bo * NC + c) * 2;
        inits[idx + 0] = syn;
        inits[idx + 1] = mem;
        float ls = summ[idx + 0], lm = summ[idx + 1];
        float nsyn = aL * syn + ls;
        float nmem = uL * syn + bL * mem + lm;
        syn = nsyn; mem = nmem;
    }
}

// ---------------- Pass 2c: re-scan from true initial state; contiguous float2 out ---
__global__ void snn_chunk_final(const float* __restrict__ curr,
                                const float* __restrict__ bias,
                                const float* __restrict__ alpha,
                                const float* __restrict__ beta,
                                const float* __restrict__ inits,
                                float* __restrict__ out)     // [B, T, O]
{
    int tid = blockIdx.x * blockDim.x + threadIdx.x;
    if (tid >= B_ * NC) return;
    int c = tid % NC;
    int b = tid / NC;
    float a0 = clamp01(alpha[0]), a1 = clamp01(alpha[1]);
    float b0 = clamp01(beta[0]),  b1 = clamp01(beta[1]);
    float s0 = bias[0],           s1 = bias[1];
    int base0 = ((b * O_ + 0) * NC + c) * 2;
    int base1 = ((b * O_ + 1) * NC + c) * 2;
    float syn0 = inits[base0 + 0], mem0 = inits[base0 + 1];
    float syn1 = inits[base1 + 0], mem1 = inits[base1 + 1];
    const float4* r0 = (const float4*)&curr[((size_t)b * O_ + 0) * T_ + (size_t)c * CH];
    const float4* r1 = (const float4*)&curr[((size_t)b * O_ + 1) * T_ + (size_t)c * CH];
    float2* __restrict__ wr = (float2*)out + ((size_t)b * T_ + (size_t)c * CH);
    for (int j = 0; j < CH / 4; ++j) {
        float4 c0 = r0[j], c1 = r1[j];
        float2 q;
        syn0 = a0 * syn0 + (c0.x + s0); mem0 = b0 * mem0 + syn0;
        syn1 = a1 * syn1 + (c1.x + s1); mem1 = b1 * mem1 + syn1;
        q.x = mem0; q.y = mem1; wr[4 * j + 0] = q;
        syn0 = a0 * syn0 + (c0.y + s0); mem0 = b0 * mem0 + syn0;
        syn1 = a1 * syn1 + (c1.y + s1); mem1 = b1 * mem1 + syn1;
        q.x = mem0; q.y = mem1; wr[4 * j + 1] = q;
        syn0 = a0 * syn0 + (c0.z + s0); mem0 = b0 * mem0 + syn0;
        syn1 = a1 * syn1 + (c1.z + s1); mem1 = b1 * mem1 + syn1;
        q.x = mem0; q.y = mem1; wr[4 * j + 2] = q;
        syn0 = a0 * syn0 + (c0.w + s0); mem0 = b0 * mem0 + syn0;
        syn1 = a1 * syn1 + (c1.w + s1); mem1 = b1 * mem1 + syn1;
        q.x = mem0; q.y = mem1; wr[4 * j + 3] = q;
    }
}

extern "C" void kernel_launch(void* const* d_in, const int* in_sizes, int n_in,
                              void* d_out, int out_size, void* d_ws, size_t ws_size,
                              hipStream_t stream) {
    const float* inp   = (const float*)d_in[0];  // [B, I, T]
    const float* W     = (const float*)d_in[1];  // [O, I]
    const float* bias  = (const float*)d_in[2];  // [O]
    const float* alpha = (const float*)d_in[3];  // [O]
    const float* beta  = (const float*)d_in[4];  // [O]
    float* out = (float*)d_out;

    // Workspace layout (floats): curr [B*O*T] | summ [B*O*NC*2] | inits [B*O*NC*2]
    float* curr  = (float*)d_ws;
    float* summ  = curr + (size_t)B_ * O_ * T_;
    float* inits = summ + (size_t)B_ * O_ * NC * 2;

    // Pass 1: WMMA GEMM. Block = 128 threads (4 waves); each wave one 16-t tile.
    dim3 g1(B_, (NT + 3) / 4);
    snn_gemm_wmma<<<g1, 128, 0, stream>>>(inp, W, curr);

    // Pass 2: chunk-parallel linear scan.
    int n2 = B_ * NC;
    snn_chunk_local<<<(n2 + 255) / 256, 256, 0, stream>>>(curr, bias, alpha, beta, summ);
    snn_chunk_scan<<<(B_ * O_ + 255) / 256, 256, 0, stream>>>(summ, alpha, beta, inits);
    snn_chunk_final<<<(n2 + 255) / 256, 256, 0, stream>>>(curr, bias, alpha, beta, inits, out);
}